// BinaryQuantumClassifier_3358664426249
// MI455X (gfx1250) — compile-verified
//
#include <hip/hip_runtime.h>
#include <math.h>

// ---------------------------------------------------------------------------
// Fused BinaryQuantumClassifier for MI455X (gfx1250, wave32)
//   compressed = x @ W_ctq^T + b_ctq        (WMMA f32 16x16x4, fp32-exact)
//   z          = 4-qubit circuit per sample (1 sample per lane, regs only)
//   out        = mean_reuse(z) @ W_cls^T + b_cls
// Memory-bound on reading x (128 MB @ 23.3 TB/s ~= 5.5 us): single fused pass.
// Fast hardware trig (v_sin_f32/v_cos_f32) -- args are O(1), no Payne-Hanek.
// ---------------------------------------------------------------------------

typedef __attribute__((ext_vector_type(2))) float v2f;
typedef __attribute__((ext_vector_type(8))) float v8f;

#define DQ 512
#define TILE_M 16
#define WAVES_PER_BLOCK 8
#define BLOCK_THREADS (WAVES_PER_BLOCK * 32)

__global__ __launch_bounds__(BLOCK_THREADS)
void bqc_fused_kernel(const float* __restrict__ x,
                      const float* __restrict__ W_ctq,
                      const float* __restrict__ b_ctq,
                      const float* __restrict__ qp,
                      const float* __restrict__ W_cls,
                      const float* __restrict__ b_cls,
                      float* __restrict__ out)
{
    // B-matrix staged pair-packed: Wlds[((k/2)*16 + n)*2 + {0,1}] = W^T[k..k+1][n]
    // (n padded 8->16 with zeros).  32 KB.
    __shared__ float Wlds[(DQ / 2) * 16 * 2];
    // per-wave scratch: C tile (16x16 f32), later reused for z.  8 KB.
    __shared__ float Scr[WAVES_PER_BLOCK][TILE_M * 16];

    const int tid  = threadIdx.x;
    const int lane = tid & 31;
    const int wave = tid >> 5;

    // ---- stage W_ctq^T (padded, pair-packed) into LDS ----
    for (int i = tid; i < (DQ / 2) * 16; i += BLOCK_THREADS) {
        const int k2 = i >> 4;          // pair index: rows k=2*k2, 2*k2+1
        const int n  = i & 15;
        float w0 = 0.0f, w1 = 0.0f;
        if (n < 8) {
            w0 = W_ctq[n * DQ + 2 * k2 + 0];
            w1 = W_ctq[n * DQ + 2 * k2 + 1];
        }
        Wlds[i * 2 + 0] = w0;
        Wlds[i * 2 + 1] = w1;
    }
    __syncthreads();

    const int tile    = blockIdx.x * WAVES_PER_BLOCK + wave;
    const int rowBase = tile * TILE_M;

    // ---- GEMM1: 16x16 tile of compressed = x @ W^T via V_WMMA_F32_16X16X4_F32
    // A 16x4 f32 layout: lane L holds M=L&15; lanes 0-15 -> K={0,1}, 16-31 -> K={2,3}
    const int   halfK = lane >> 4;             // 0 or 1
    const int   bCol  = lane & 15;             // N for B/C fragments
    const float* aPtr = x + (size_t)(rowBase + (lane & 15)) * DQ + 2 * halfK;

    v8f acc = {0.f, 0.f, 0.f, 0.f, 0.f, 0.f, 0.f, 0.f};
    for (int k0 = 0; k0 < DQ; k0 += 4) {
        __builtin_prefetch(aPtr + k0 + 32, 0, 1);            // global_prefetch_b8
        v2f a = *(const v2f*)(aPtr + k0);                    // K = k0+2h, k0+2h+1
        v2f b = *(const v2f*)&Wlds[(((k0 >> 1) + halfK) * 16 + bCol) * 2];
        acc = __builtin_amdgcn_wmma_f32_16x16x4_f32(
            false, a, false, b, (short)0, acc, false, false);
    }

    // ---- spill C tile to per-wave LDS (C layout: VGPR r -> M=r+8*halfK, N=bCol)
    float* C = Scr[wave];
    #pragma unroll
    for (int r = 0; r < 8; r++)
        C[(r + 8 * halfK) * 16 + bCol] = acc[r];
    __syncthreads();

    // ---- quantum circuit: lane L simulates sample (local row L>>1, reuse L&1)
    const int lrow = lane >> 1;
    const int rsel = lane & 1;
    float feat[4];
    #pragma unroll
    for (int i = 0; i < 4; i++)
        feat[i] = C[lrow * 16 + rsel * 4 + i] + b_ctq[rsel * 4 + i];

    float sr[16], si[16];
    #pragma unroll
    for (int k = 0; k < 16; k++) { sr[k] = 0.0f; si[k] = 0.0f; }
    sr[0] = 1.0f;

    const float INVSQRT2 = 0.70710678118654752440f;
    const float HALF_PI  = 1.57079632679489661923f;

    // one fused 2x2 complex unitary per wire: U = RZ(g)*RY(b)*RX(a)*RY(th_f)*H
    #pragma unroll
    for (int w = 0; w < 4; w++) {
        const float a2 = 0.5f * qp[3 * w + 0];
        const float b2 = 0.5f * qp[3 * w + 1];
        const float g2 = 0.5f * qp[3 * w + 2];
        const float ca = __cosf(a2), sa = __sinf(a2);
        const float cb = __cosf(b2), sb = __sinf(b2);
        const float cg = __cosf(g2), sg = __sinf(g2);
        // M = RZ * RY * RX
        const float m00r =  cb * ca, m00i =  sb * sa;
        const float m01r = -sb * ca, m01i = -cb * sa;
        const float m10r =  sb * ca, m10i = -cb * sa;
        const float m11r =  cb * ca, m11i = -sb * sa;
        const float M00r = m00r * cg + m00i * sg, M00i = m00i * cg - m00r * sg;
        const float M01r = m01r * cg + m01i * sg, M01i = m01i * cg - m01r * sg;
        const float M10r = m10r * cg - m10i * sg, M10i = m10i * cg + m10r * sg;
        const float M11r = m11r * cg - m11i * sg, M11i = m11i * cg + m11r * sg;
        // G = RY(feat*pi) * H  (real)
        const float tf2 = feat[w] * HALF_PI;     // (feat*pi)/2
        const float cf = __cosf(tf2), sf = __sinf(tf2);
        const float g00 = INVSQRT2 * (cf - sf);
        const float g01 = INVSQRT2 * (cf + sf);
        const float g10 = g01, g11 = -g00;
        // U = M * G
        const float u00r = M00r * g00 + M01r * g10, u00i = M00i * g00 + M01i * g10;
        const float u01r = M00r * g01 + M01r * g11, u01i = M00i * g01 + M01i * g11;
        const float u10r = M10r * g00 + M11r * g10, u10i = M10i * g00 + M11i * g10;
        const float u11r = M10r * g01 + M11r * g11, u11i = M10i * g01 + M11i * g11;

        const int bit = 1 << (3 - w);
        #pragma unroll
        for (int idx = 0; idx < 16; idx++) {
            if (idx & bit) continue;
            const int j = idx, k = idx | bit;
            const float a0r = sr[j], a0i = si[j], a1r = sr[k], a1i = si[k];
            sr[j] = u00r * a0r - u00i * a0i + u01r * a1r - u01i * a1i;
            si[j] = u00r * a0i + u00i * a0r + u01r * a1i + u01i * a1r;
            sr[k] = u10r * a0r - u10i * a0i + u11r * a1r - u11i * a1i;
            si[k] = u10r * a0i + u10i * a0r + u11r * a1i + u11i * a1r;
        }
    }

    // ---- CNOT ring (0,1)(1,2)(2,3)(3,0): pure register swaps
    const int cn[4][2] = {{0, 1}, {1, 2}, {2, 3}, {3, 0}};
    #pragma unroll
    for (int g = 0; g < 4; g++) {
        const int bc = 1 << (3 - cn[g][0]);
        const int bt = 1 << (3 - cn[g][1]);
        #pragma unroll
        for (int idx = 0; idx < 16; idx++) {
            if ((idx & bc) && !(idx & bt)) {
                const int j = idx, k = idx | bt;
                float tr = sr[j]; sr[j] = sr[k]; sr[k] = tr;
                float ti = si[j]; si[j] = si[k]; si[k] = ti;
            }
        }
    }

    // ---- <Z_i> marginals
    float pr[16];
    #pragma unroll
    for (int idx = 0; idx < 16; idx++)
        pr[idx] = sr[idx] * sr[idx] + si[idx] * si[idx];
    float z[4];
    #pragma unroll
    for (int w = 0; w < 4; w++) {
        const int bit = 1 << (3 - w);
        float zz = 0.0f;
        #pragma unroll
        for (int idx = 0; idx < 16; idx++)
            zz += (idx & bit) ? -pr[idx] : pr[idx];
        z[w] = zz;
    }

    // ---- mean over reuse + classifier head (lane -> (row = lane>>1, cls = lane&1))
    float* Z = Scr[wave];                 // reuse scratch (same-wave DS is in-order)
    #pragma unroll
    for (int i = 0; i < 4; i++)
        Z[lane * 4 + i] = z[i];
    __syncthreads();

    const int orow = lane >> 1;
    const int ocls = lane & 1;
    float o = b_cls[ocls];
    #pragma unroll
    for (int i = 0; i < 4; i++) {
        const float zm = 0.5f * (Z[(2 * orow + 0) * 4 + i] + Z[(2 * orow + 1) * 4 + i]);
        o += zm * W_cls[ocls * 4 + i];
    }
    out[(size_t)(rowBase + orow) * 2 + ocls] = o;
}

extern "C" void kernel_launch(void* const* d_in, const int* in_sizes, int n_in,
                              void* d_out, int out_size, void* d_ws, size_t ws_size,
                              hipStream_t stream) {
    const float* x     = (const float*)d_in[0];
    const float* W_ctq = (const float*)d_in[1];
    const float* b_ctq = (const float*)d_in[2];
    const float* qpar  = (const float*)d_in[3];
    const float* W_cls = (const float*)d_in[4];
    const float* b_cls = (const float*)d_in[5];
    float* out = (float*)d_out;

    const int B      = in_sizes[0] / DQ;                 // 65536
    const int tiles  = B / TILE_M;                       // 4096 waves
    const int blocks = tiles / WAVES_PER_BLOCK;          // 512 blocks of 256 thr

    bqc_fused_kernel<<<blocks, BLOCK_THREADS, 0, stream>>>(
        x, W_ctq, b_ctq, qpar, W_cls, b_cls, out);
}